// Attention_25855703122201
// MI455X (gfx1250) — compile-verified
//
#include <hip/hip_runtime.h>
#include <math.h>

// ---- problem constants (match reference) ----
#define NUM_SEQS      32
#define NUM_HEADS     32
#define NUM_KV_HEADS  8
#define GQA           4        // NUM_HEADS / NUM_KV_HEADS
#define HEAD          128
#define BS            16       // tokens per paged block
#define MAXB          128      // max blocks per seq
#define WAVES         8        // waves per workgroup (256 threads, wave32)
#define PARTS         4        // split-K partitions per (seq, kv_head)

typedef __attribute__((ext_vector_type(2))) float v2f;
typedef __attribute__((ext_vector_type(8))) float v8f;

// Flash-decode paged attention, fp32, WMMA f32 16x16x4 (wave32).
//   QK: D_s[m=key, n=head]   A=K-tile(16x4), B=Q(4x16, heads 0..3 live, rest 0)
//   PV: D_o[m=dim, n=head]   A=V^T(16x4),    B=P(4x16)
// Layouts (ISA 7.12.2, 32-bit, wave32):
//   A 16x4 : elem(M=lane%16, K=r+2*(lane>=16)) in VGPR r
//   B 4x16 : elem(K=r+2*(lane>=16), N=lane%16) in VGPR r
//   C/D    : elem(M=r+8*(lane>=16), N=lane%16) in VGPR r
//
// FINAL=true : single pass, normalized result straight to `out`.
// FINAL=false: split-K partial -> ws_acc (unnormalized, rescaled to partition
//              max) + ws_ml (per-head m, l); merged by pa_combine.
template <int NPART, bool FINAL>
__global__ __launch_bounds__(256) void pa_decode_f32_wmma(
    const float* __restrict__ q,      // [32, 32*128]
    const float* __restrict__ kc,     // [4096, 16, 8, 128]
    const float* __restrict__ vc,     // [4096, 16, 8, 128]
    const int*   __restrict__ bt,     // [32, 128]
    const int*   __restrict__ sl,     // [32]
    float*       __restrict__ out,    // [32, 32*128]        (FINAL)
    float*       __restrict__ ws_acc, // [256*NPART, 512]    (!FINAL)
    float*       __restrict__ ws_ml)  // [256*NPART, 8] m[4],l[4]
{
    const int pi   = blockIdx.x;          // partition index
    const int sh   = pi / NPART;          // (seq, kv_head) index
    const int part = pi - sh * NPART;
    const int s    = sh >> 3;
    const int h    = sh & 7;
    const int tid  = threadIdx.x;
    const int wave = tid >> 5;
    const int lane = tid & 31;
    const int hl   = lane & 15;   // N coord (head) / M coord (key or dim)
    const int hi   = lane >> 4;   // lane half -> K-dim split / M upper half

    __shared__ float lm[WAVES][16];
    __shared__ float ll[WAVES][16];
    __shared__ float lo_acc[GQA * HEAD];   // 512 floats

    lo_acc[tid]       = 0.0f;              // zero combine buffer
    lo_acc[tid + 256] = 0.0f;

    const int len = sl[s];
    const int nb  = (len + BS - 1) >> 4;
    const int b0  = part * (MAXB / NPART);
    const int b1  = min(nb, b0 + (MAXB / NPART));

    // SCALE * log2(e): softmax done in exp2 domain
    const float csc = 0.08838834764831845f * 1.4426950408889634f;

    // ---- preload Q in B-operand layout: qb[c] covers dims [4c, 4c+4) ----
    v2f qb[32];
    {
        const float* qp = q + (size_t)s * (NUM_HEADS * HEAD)
                            + (size_t)(h * GQA + hl) * HEAD;
        #pragma unroll
        for (int c = 0; c < 32; ++c) {
            v2f v = {0.0f, 0.0f};
            if (hl < GQA) v = *(const v2f*)(qp + 4 * c + 2 * hi);
            qb[c] = v;
        }
    }

    v8f acc[8] = {};                  // D_o accumulators, 8 dim-chunks of 16
    float mrun = -INFINITY;
    float lrun = 0.0f;

    for (int b = b0 + wave; b < b1; b += WAVES) {
        const int blk = bt[s * MAXB + b];
        const size_t base = (size_t)blk * (BS * NUM_KV_HEADS * HEAD) + (size_t)h * HEAD;

        // ---- QK^T: two independent WMMA chains for ILP ----
        const float* kp = kc + base + (size_t)hl * (NUM_KV_HEADS * HEAD);
        v8f st0 = {};
        v8f st1 = {};
        #pragma unroll
        for (int c = 0; c < 32; c += 2) {
            v2f a0 = *(const v2f*)(kp + 4 * c + 2 * hi);
            v2f a1 = *(const v2f*)(kp + 4 * (c + 1) + 2 * hi);
            st0 = __builtin_amdgcn_wmma_f32_16x16x4_f32(
                      false, a0, false, qb[c], (short)0, st0, false, false);
            st1 = __builtin_amdgcn_wmma_f32_16x16x4_f32(
                      false, a1, false, qb[c + 1], (short)0, st1, false, false);
        }

        // ---- online softmax (per head column = per lane) ----
        const int kb = b * BS + 8 * hi;       // key index of row r in this half
        float u[8];
        #pragma unroll
        for (int r = 0; r < 8; ++r) {
            const float uu = (st0[r] + st1[r]) * csc;
            u[r] = (kb + r < len) ? uu : -INFINITY;
        }
        float mb = u[0];
        #pragma unroll
        for (int r = 1; r < 8; ++r) mb = fmaxf(mb, u[r]);
        mb = fmaxf(mb, __shfl_xor(mb, 16, 32));          // combine lane halves

        const float mnew = fmaxf(mrun, mb);
        const float f    = exp2f(mrun - mnew);           // exp2(-inf)=0 first block
        mrun = mnew;

        float pr[8], ps = 0.0f;
        #pragma unroll
        for (int r = 0; r < 8; ++r) { pr[r] = exp2f(u[r] - mnew); ps += pr[r]; }
        ps += __shfl_xor(ps, 16, 32);
        lrun = lrun * f + ps;

        #pragma unroll
        for (int c = 0; c < 8; ++c)
            #pragma unroll
            for (int r = 0; r < 8; ++r) acc[c][r] *= f;

        // ---- relayout P: D-tile -> B-operand chunks (keys 4j..4j+3) ----
        float t[8];
        #pragma unroll
        for (int r = 0; r < 8; ++r) t[r] = __shfl_xor(pr[r], 16, 32);
        const bool lo16 = (lane < 16);
        v2f pb[4];
        pb[0].x = lo16 ? pr[0] : t[2];  pb[0].y = lo16 ? pr[1] : t[3];  // keys 0..3
        pb[1].x = lo16 ? pr[4] : t[6];  pb[1].y = lo16 ? pr[5] : t[7];  // keys 4..7
        pb[2].x = lo16 ? t[0] : pr[2];  pb[2].y = lo16 ? t[1] : pr[3];  // keys 8..11
        pb[3].x = lo16 ? t[4] : pr[6];  pb[3].y = lo16 ? t[5] : pr[7];  // keys 12..15

        // ---- PV: out[dim, head] += V^T x P (8 independent chains) ----
        const float* vp = vc + base + hl;   // dim column = 16c + lane%16
        #pragma unroll
        for (int c = 0; c < 8; ++c) {
            #pragma unroll
            for (int j = 0; j < 4; ++j) {
                v2f a;
                a.x = vp[(size_t)(4 * j + 0 + 2 * hi) * (NUM_KV_HEADS * HEAD) + 16 * c];
                a.y = vp[(size_t)(4 * j + 1 + 2 * hi) * (NUM_KV_HEADS * HEAD) + 16 * c];
                acc[c] = __builtin_amdgcn_wmma_f32_16x16x4_f32(
                             false, a, false, pb[j], (short)0, acc[c], false, false);
            }
        }
    }

    // ---- cross-wave flash combine via LDS ----
    if (lane < 16) { lm[wave][lane] = mrun; ll[wave][lane] = lrun; }
    __syncthreads();

    float mg = -INFINITY;
    #pragma unroll
    for (int w = 0; w < WAVES; ++w) mg = fmaxf(mg, lm[w][hl]);
    float L = 0.0f;
    #pragma unroll
    for (int w = 0; w < WAVES; ++w) L += ll[w][hl] * exp2f(lm[w][hl] - mg);

    float scale = exp2f(mrun - mg);        // rescale this wave's acc to mg
    if (!(mg > -INFINITY)) scale = 0.0f;   // empty partition: avoid exp2(NaN)
    if (FINAL) scale = scale / L;          // single pass: normalize here

    if (hl < GQA) {
        #pragma unroll
        for (int c = 0; c < 8; ++c)
            #pragma unroll
            for (int r = 0; r < 8; ++r)
                atomicAdd(&lo_acc[hl * HEAD + 16 * c + r + 8 * hi], acc[c][r] * scale);
    }
    __syncthreads();

    if (FINAL) {
        if (tid < 128) {
            const float4 v = ((const float4*)lo_acc)[tid];
            ((float4*)(out + (size_t)s * (NUM_HEADS * HEAD)
                           + (size_t)h * (GQA * HEAD)))[tid] = v;
        }
    } else {
        if (tid < 128) {
            const float4 v = ((const float4*)lo_acc)[tid];
            ((float4*)(ws_acc + (size_t)pi * (GQA * HEAD)))[tid] = v;
        }
        if (tid < GQA) {                   // per-head partition stats
            ws_ml[pi * 8 + tid]       = mg;
            ws_ml[pi * 8 + 4 + tid]   = L;
        }
    }
}

// Merge NPART split-K partials: out = sum_p acc_p*exp2(m_p-gm) / sum_p l_p*exp2(m_p-gm)
__global__ __launch_bounds__(128) void pa_combine(
    const float* __restrict__ ws_acc,
    const float* __restrict__ ws_ml,
    float*       __restrict__ out)
{
    const int sh   = blockIdx.x;           // (seq, kv_head)
    const int tid  = threadIdx.x;          // 128 threads x float4 = 512 floats
    const int head = tid >> 5;             // (tid*4)/128

    float m[PARTS];
    #pragma unroll
    for (int p = 0; p < PARTS; ++p) m[p] = ws_ml[(sh * PARTS + p) * 8 + head];

    float gm = m[0];
    #pragma unroll
    for (int p = 1; p < PARTS; ++p) gm = fmaxf(gm, m[p]);

    float fac[PARTS];
    float Lt = 0.0f;
    #pragma unroll
    for (int p = 0; p < PARTS; ++p) {
        float f = (m[p] > -INFINITY) ? exp2f(m[p] - gm) : 0.0f;
        fac[p] = f;
        Lt += ws_ml[(sh * PARTS + p) * 8 + 4 + head] * f;
    }
    const float inv = 1.0f / Lt;

    float ax = 0.f, ay = 0.f, az = 0.f, aw = 0.f;
    #pragma unroll
    for (int p = 0; p < PARTS; ++p) {
        const float4 a =
            ((const float4*)(ws_acc + (size_t)(sh * PARTS + p) * (GQA * HEAD)))[tid];
        ax += a.x * fac[p]; ay += a.y * fac[p];
        az += a.z * fac[p]; aw += a.w * fac[p];
    }
    float4 r; r.x = ax * inv; r.y = ay * inv; r.z = az * inv; r.w = aw * inv;
    const int s = sh >> 3, h = sh & 7;
    ((float4*)(out + (size_t)s * (NUM_HEADS * HEAD) + (size_t)h * (GQA * HEAD)))[tid] = r;
}

extern "C" void kernel_launch(void* const* d_in, const int* in_sizes, int n_in,
                              void* d_out, int out_size, void* d_ws, size_t ws_size,
                              hipStream_t stream) {
    (void)in_sizes; (void)n_in; (void)out_size;
    const float* q  = (const float*)d_in[0];
    const float* kc = (const float*)d_in[1];
    const float* vc = (const float*)d_in[2];
    const int*   bt = (const int*)d_in[3];
    const int*   sl = (const int*)d_in[4];
    float*       o  = (float*)d_out;

    const int nsh = NUM_SEQS * NUM_KV_HEADS;                 // 256
    const size_t need = (size_t)nsh * PARTS * (GQA * HEAD + 8) * sizeof(float); // ~2.1MB

    if (ws_size >= need) {
        // split-K: 1024 WGs of 8 waves -> 4x streams, 4x shorter tail
        float* ws_acc = (float*)d_ws;
        float* ws_ml  = ws_acc + (size_t)nsh * PARTS * (GQA * HEAD);
        hipLaunchKernelGGL((pa_decode_f32_wmma<PARTS, false>),
                           dim3(nsh * PARTS), dim3(256), 0, stream,
                           q, kc, vc, bt, sl, (float*)nullptr, ws_acc, ws_ml);
        hipLaunchKernelGGL(pa_combine, dim3(nsh), dim3(128), 0, stream,
                           ws_acc, ws_ml, o);
    } else {
        // fallback: single-pass, one WG per (seq, kv_head)
        hipLaunchKernelGGL((pa_decode_f32_wmma<1, true>),
                           dim3(nsh), dim3(256), 0, stream,
                           q, kc, vc, bt, sl, o, (float*)nullptr, (float*)nullptr);
    }
}